// GCNNet_9156870275402
// MI455X (gfx1250) — compile-verified
//
#include <hip/hip_runtime.h>
#include <math.h>

typedef __attribute__((ext_vector_type(2))) float v2f;
typedef __attribute__((ext_vector_type(8))) float v8f;

#define N_NODES 50000
#define N_EDGES 800000
#define D 128
#define ND (N_NODES * D)   // 6,400,000

// ---------------------------------------------------------------------------
// Degree: deg[i] = 1 (self loop) + #edges with dst == i
// ---------------------------------------------------------------------------
__global__ void init_deg_kernel(float* __restrict__ deg) {
    int i = blockIdx.x * blockDim.x + threadIdx.x;
    if (i < N_NODES) deg[i] = 1.0f;
}

__global__ void count_deg_kernel(const int* __restrict__ dst, float* __restrict__ deg) {
    int e = blockIdx.x * blockDim.x + threadIdx.x;
    if (e < N_EDGES) atomicAdd(&deg[dst[e]], 1.0f);
}

__global__ void rsqrt_kernel(float* __restrict__ deg) {
    int i = blockIdx.x * blockDim.x + threadIdx.x;
    if (i < N_NODES) deg[i] = rsqrtf(deg[i]);
}

// ---------------------------------------------------------------------------
// H = X @ W, scaled by dinv[row]:  hs[r, :] = dinv[r] * (X @ W)[r, :]
// fp32 WMMA 16x16x4. One wave per 16x16 output tile; 8 waves/block cover one
// 16-row strip across all 128 output columns. 50000 = 3125 * 16, no tail.
//
// A 16x4 f32 (2 VGPRs/lane): lanes 0-15 -> M=lane, v0=K0 v1=K1;
//                            lanes 16-31 -> M=lane-16, v0=K2 v1=K3.
// B 4x16 f32 (2 VGPRs/lane): lanes 0-15 -> N=lane, v0=K0 v1=K1;
//                            lanes 16-31 -> N=lane-16, v0=K2 v1=K3.
// C/D 16x16 f32 (8 VGPRs): VGPR r, lanes 0-15 -> M=r, lanes 16-31 -> M=r+8.
// ---------------------------------------------------------------------------
__global__ __launch_bounds__(256)
void gemm_scale_kernel(const float* __restrict__ X, const float* __restrict__ W,
                       const float* __restrict__ dinv, float* __restrict__ hs) {
    const int lane = threadIdx.x & 31;
    const int wave = threadIdx.x >> 5;         // 0..7 -> 16-col tile
    const int row0 = blockIdx.x * 16;
    const int col0 = wave * 16;
    const int mn   = lane & 15;                // M for A, N for B
    const int kb   = (lane >> 4) << 1;         // 0 or 2: lane-half K offset

    const float* arow = X + (size_t)(row0 + mn) * D;

    v8f c = {};
    #pragma unroll
    for (int ks = 0; ks < D; ks += 4) {
        v2f a = *(const v2f*)(arow + ks + kb);               // K = ks+kb, ks+kb+1
        v2f b;
        b.x = W[(size_t)(ks + kb)     * D + col0 + mn];
        b.y = W[(size_t)(ks + kb + 1) * D + col0 + mn];
        c = __builtin_amdgcn_wmma_f32_16x16x4_f32(
                /*neg_a=*/false, a, /*neg_b=*/false, b,
                /*c_mod=*/(short)0, c, /*reuse_a=*/false, /*reuse_b=*/false);
    }

    const int rbase = (lane >> 4) << 3;        // lanes 16-31 hold M = r+8
    #pragma unroll
    for (int r = 0; r < 8; ++r) {
        int row = row0 + rbase + r;
        hs[(size_t)row * D + col0 + mn] = c[r] * dinv[row];
    }
}

// ---------------------------------------------------------------------------
// Edge scatter: agg[dst, :] += hs[src, :]   (one wave per edge, 4 f32 atomics
// per lane; accumulator is L2-resident -> atomics execute at L2)
// ---------------------------------------------------------------------------
__global__ __launch_bounds__(256)
void scatter_kernel(const float* __restrict__ hs, const int* __restrict__ src,
                    const int* __restrict__ dst, float* __restrict__ agg) {
    int e = blockIdx.x * 8 + (threadIdx.x >> 5);
    if (e >= N_EDGES) return;
    int lane = threadIdx.x & 31;
    const float* h = hs  + (size_t)src[e] * D;
    float*       a = agg + (size_t)dst[e] * D;
    #pragma unroll
    for (int t = 0; t < 4; ++t)
        atomicAdd(a + lane + 32 * t, h[lane + 32 * t]);
}

// ---------------------------------------------------------------------------
// Epilogue: out = [elu](dinv[row] * agg + bias)
// ---------------------------------------------------------------------------
__global__ __launch_bounds__(256)
void epilogue_kernel(const float* __restrict__ agg, const float* __restrict__ dinv,
                     const float* __restrict__ bias, float* __restrict__ out,
                     int apply_elu) {
    unsigned idx = blockIdx.x * blockDim.x + threadIdx.x;   // < ND exactly
    int row = idx >> 7;
    int f   = idx & 127;
    float v = agg[idx] * dinv[row] + bias[f];
    if (apply_elu) v = (v > 0.0f) ? v : expm1f(v);
    out[idx] = v;
}

// ---------------------------------------------------------------------------
extern "C" void kernel_launch(void* const* d_in, const int* in_sizes, int n_in,
                              void* d_out, int out_size, void* d_ws, size_t ws_size,
                              hipStream_t stream) {
    const float* X   = (const float*)d_in[0];          // [50000,128]
    const int*   ei  = (const int*)  d_in[1];          // [2,800000] row-major
    const float* W1  = (const float*)d_in[2];          // [128,128]
    const float* b1  = (const float*)d_in[3];          // [128]
    const float* W2  = (const float*)d_in[4];          // [128,128]
    const float* b2  = (const float*)d_in[5];          // [128]
    float*       out = (float*)d_out;                  // [50000,128]

    const int* src = ei;                               // edge_index[0]
    const int* dst = ei + N_EDGES;                     // edge_index[1]

    // Workspace layout (floats): dinv[50000 pad->50176] | hs[ND] | agg[ND]
    float* dinv = (float*)d_ws;
    float* hs   = dinv + 50176;
    float* agg  = hs + ND;
    float* x1   = out;                                 // layer-1 activations live in d_out

    const int blk = 256;
    const int gNodes = (N_NODES + blk - 1) / blk;      // 196
    const int gEdges = (N_EDGES + blk - 1) / blk;      // 3125
    const int gGemm  = N_NODES / 16;                   // 3125
    const int gScat  = (N_EDGES + 7) / 8;              // 100000
    const int gElem  = ND / blk;                       // 25000

    // Degree / normalization (shared by both layers)
    init_deg_kernel<<<gNodes, blk, 0, stream>>>(dinv);
    count_deg_kernel<<<gEdges, blk, 0, stream>>>(dst, dinv);
    rsqrt_kernel<<<gNodes, blk, 0, stream>>>(dinv);

    // ---- Layer 1 ----
    gemm_scale_kernel<<<gGemm, blk, 0, stream>>>(X, W1, dinv, hs);
    hipMemcpyAsync(agg, hs, (size_t)ND * sizeof(float),
                   hipMemcpyDeviceToDevice, stream);   // self-loop term
    scatter_kernel<<<gScat, blk, 0, stream>>>(hs, src, dst, agg);
    epilogue_kernel<<<gElem, blk, 0, stream>>>(agg, dinv, b1, x1, /*elu=*/1);

    // ---- Layer 2 ----
    gemm_scale_kernel<<<gGemm, blk, 0, stream>>>(x1, W2, dinv, hs);
    hipMemcpyAsync(agg, hs, (size_t)ND * sizeof(float),
                   hipMemcpyDeviceToDevice, stream);
    scatter_kernel<<<gScat, blk, 0, stream>>>(hs, src, dst, agg);
    epilogue_kernel<<<gElem, blk, 0, stream>>>(agg, dinv, b2, out, /*elu=*/0);
}